// ModuleSoftsplat_77833397338457
// MI455X (gfx1250) — compile-verified
//
#include <hip/hip_runtime.h>

// Problem constants (from the reference): B=4, C=64, H=288, W=512
constexpr int Bn = 4;
constexpr int Cn = 64;
constexpr int Hn = 288;
constexpr int Wn = 512;
constexpr int HW = Hn * Wn;          // 147456
constexpr int NPIX = Bn * HW;        // 589824 source pixels
constexpr int NOUT = Bn * Cn * HW;   // 37748736 output elements

// ---------------------------------------------------------------------------
// Zero-fill with 128-bit stores (graph-capture safe; no hipMemset)
// ---------------------------------------------------------------------------
__global__ void __launch_bounds__(256) zero_f4(float4* __restrict__ p, int n4) {
    int i = blockIdx.x * blockDim.x + threadIdx.x;
    if (i < n4) p[i] = make_float4(0.f, 0.f, 0.f, 0.f);
}

// ---------------------------------------------------------------------------
// Forward bilinear splat. One thread per source pixel.
//   input  [B][C][H*W]   flow [B][2][H*W]   metric [B][1][H*W]
//   out    [B][C][H*W]   (numerator accumulator, == final output buffer)
//   norm   [B][H*W]      (denominator accumulator, lives in d_ws)
// Accumulation surface is ~153 MB -> fits MI455X's 192 MB L2, so the
// global_atomic_add_f32 scatter resolves in L2, not HBM.
// ---------------------------------------------------------------------------
__global__ void __launch_bounds__(256) splat_kernel(
    const float* __restrict__ input,
    const float* __restrict__ flow,
    const float* __restrict__ metric,
    float* __restrict__ out,
    float* __restrict__ norm)
{
    int p = blockIdx.x * blockDim.x + threadIdx.x;
    if (p >= NPIX) return;

    int b   = p / HW;
    int pix = p - b * HW;
    int y   = pix / Wn;
    int x   = pix - y * Wn;

    // Per-pixel scalars (coalesced loads across the wave)
    float fdx = flow[(size_t)(b * 2 + 0) * HW + pix];
    float fdy = flow[(size_t)(b * 2 + 1) * HW + pix];
    float m   = expf(metric[(size_t)b * HW + pix]);

    float fx = (float)x + fdx;
    float fy = (float)y + fdy;
    float x0f = floorf(fx);
    float y0f = floorf(fy);
    int x0 = (int)x0f, y0 = (int)y0f;
    int x1 = x0 + 1,   y1 = y0 + 1;

    float ax = fx - x0f;            // == fx - x0; (x1 - fx) == 1 - ax
    float ay = fy - y0f;
    float bx = 1.f - ax;
    float by = 1.f - ay;

    float w00 = bx * by;            // (x0,y0)
    float w10 = ax * by;            // (x1,y0)
    float w01 = bx * ay;            // (x0,y1)
    float w11 = ax * ay;            // (x1,y1)

    bool vx0 = (x0 >= 0) & (x0 < Wn);
    bool vx1 = (x1 >= 0) & (x1 < Wn);
    bool vy0 = (y0 >= 0) & (y0 < Hn);
    bool vy1 = (y1 >= 0) & (y1 < Hn);
    bool v00 = vx0 & vy0;
    bool v10 = vx1 & vy0;
    bool v01 = vx0 & vy1;
    bool v11 = vx1 & vy1;

    int i00 = y0 * Wn + x0;
    int i10 = y0 * Wn + x1;
    int i01 = y1 * Wn + x0;
    int i11 = y1 * Wn + x1;

    // Denominator channel (the reference's 65th channel)
    float* nrmB = norm + (size_t)b * HW;
    if (v00) atomicAdd(nrmB + i00, m * w00);
    if (v10) atomicAdd(nrmB + i10, m * w10);
    if (v01) atomicAdd(nrmB + i01, m * w01);
    if (v11) atomicAdd(nrmB + i11, m * w11);

    // Numerator channels: coalesced load per channel, 4 L2 atomics per channel
    const float* inB  = input + (size_t)b * Cn * HW + pix;
    float*       outB = out   + (size_t)b * Cn * HW;

    #pragma unroll 4
    for (int c = 0; c < Cn; ++c) {
        // Speculative prefetch 8 channels ahead (global_prefetch_b8);
        // OOB speculative prefetches are silently dropped by hardware.
        __builtin_prefetch(inB + (size_t)(c + 8) * HW, 0, 0);

        float v = inB[(size_t)c * HW] * m;
        float* oc = outB + (size_t)c * HW;
        if (v00) atomicAdd(oc + i00, v * w00);
        if (v10) atomicAdd(oc + i10, v * w10);
        if (v01) atomicAdd(oc + i01, v * w01);
        if (v11) atomicAdd(oc + i11, v * w11);
    }
}

// ---------------------------------------------------------------------------
// out[b][c][pix] /= (norm[b][pix] == 0 ? 1 : norm[b][pix]), vectorized x4.
// norm (2.4 MB) stays L2-resident across the 64 channel passes.
// ---------------------------------------------------------------------------
__global__ void __launch_bounds__(256) normalize4(
    float4* __restrict__ out, const float* __restrict__ norm)
{
    constexpr int HW4 = HW / 4;
    constexpr int N4  = NOUT / 4;
    int i = blockIdx.x * blockDim.x + threadIdx.x;
    if (i >= N4) return;

    int b    = i / (Cn * HW4);
    int rem  = i - b * (Cn * HW4);
    int pix4 = rem % HW4;

    float4 n = reinterpret_cast<const float4*>(norm)[(size_t)b * HW4 + pix4];
    n.x = (n.x == 0.f) ? 1.f : n.x;
    n.y = (n.y == 0.f) ? 1.f : n.y;
    n.z = (n.z == 0.f) ? 1.f : n.z;
    n.w = (n.w == 0.f) ? 1.f : n.w;

    float4 o = out[i];
    o.x /= n.x; o.y /= n.y; o.z /= n.z; o.w /= n.w;
    out[i] = o;
}

// ---------------------------------------------------------------------------
extern "C" void kernel_launch(void* const* d_in, const int* in_sizes, int n_in,
                              void* d_out, int out_size, void* d_ws, size_t ws_size,
                              hipStream_t stream)
{
    const float* input  = (const float*)d_in[0];  // [B,C,H,W]
    const float* flow   = (const float*)d_in[1];  // [B,2,H,W]
    const float* metric = (const float*)d_in[2];  // [B,1,H,W]
    float* out  = (float*)d_out;                  // [B,C,H,W]
    float* norm = (float*)d_ws;                   // [B,H,W] denominator

    constexpr int TB = 256;

    // Zero accumulators (d_out is the numerator accumulator; d_ws the denom)
    constexpr int out4 = NOUT / 4;
    constexpr int nrm4 = NPIX / 4;
    zero_f4<<<(out4 + TB - 1) / TB, TB, 0, stream>>>((float4*)out, out4);
    zero_f4<<<(nrm4 + TB - 1) / TB, TB, 0, stream>>>((float4*)norm, nrm4);

    // Scatter splat
    splat_kernel<<<(NPIX + TB - 1) / TB, TB, 0, stream>>>(input, flow, metric, out, norm);

    // In-place normalize
    normalize4<<<(out4 + TB - 1) / TB, TB, 0, stream>>>((float4*)out, norm);
}